// GaussianOnlyCompressor3D_86955907874914
// MI455X (gfx1250) — compile-verified
//
#include <hip/hip_runtime.h>

typedef __attribute__((ext_vector_type(2))) float v2f;
typedef __attribute__((ext_vector_type(8))) float v8f;

#define KG      256   // gaussians
#define CDIM    27    // SH dim
#define GSTRIDE 12    // padded per-gaussian param stride (floats)
#define TN      128   // probes per block (8 waves x 16)
#define NSTEPS  64    // KG / 4 (k-steps per WMMA_16x16x4)
#define EPSF    1e-8f

// ---------------------------------------------------------------------------
// Kernel 1: per-gaussian precompute + F swizzled into WMMA B-fragment layout
// G[k*12]  = { mux, muy, muz, r^2, p00, p01, p02, p11, p12, p22, 0, 0 }
// Bsw layout: [tile(2)][step(64)][lane(32)] -> float2 { F[k][c], F[k+1][c] }
//   with k = step*4 + 2*(lane>>4), c = tile*16 + (lane&15), zero if c >= 27
// ---------------------------------------------------------------------------
__global__ __launch_bounds__(256) void precompute_kernel(
    const float* __restrict__ mu, const float* __restrict__ log_s,
    const float* __restrict__ q, const float* __restrict__ F,
    float* __restrict__ G, float* __restrict__ Bsw)
{
    int k = threadIdx.x;  // 256 threads == KG
    {
        float qw = q[k*4+0], qx = q[k*4+1], qy = q[k*4+2], qz = q[k*4+3];
        float nrm = sqrtf(qw*qw + qx*qx + qy*qy + qz*qz);
        float inv = 1.0f / (nrm + EPSF);
        qw *= inv; qx *= inv; qy *= inv; qz *= inv;

        float r00 = 1.f - 2.f*(qy*qy + qz*qz), r01 = 2.f*(qx*qy - qw*qz), r02 = 2.f*(qx*qz + qw*qy);
        float r10 = 2.f*(qx*qy + qw*qz), r11 = 1.f - 2.f*(qx*qx + qz*qz), r12 = 2.f*(qy*qz - qw*qx);
        float r20 = 2.f*(qx*qz - qw*qy), r21 = 2.f*(qy*qz + qw*qx), r22 = 1.f - 2.f*(qx*qx + qy*qy);

        float s0 = __expf(log_s[k*3+0]);
        float s1 = __expf(log_s[k*3+1]);
        float s2 = __expf(log_s[k*3+2]);
        float i0 = 1.f/(s0*s0 + EPSF), i1 = 1.f/(s1*s1 + EPSF), i2 = 1.f/(s2*s2 + EPSF);

        float p00 = r00*r00*i0 + r01*r01*i1 + r02*r02*i2;
        float p01 = r00*r10*i0 + r01*r11*i1 + r02*r12*i2;
        float p02 = r00*r20*i0 + r01*r21*i1 + r02*r22*i2;
        float p11 = r10*r10*i0 + r11*r11*i1 + r12*r12*i2;
        float p12 = r10*r20*i0 + r11*r21*i1 + r12*r22*i2;
        float p22 = r20*r20*i0 + r21*r21*i1 + r22*r22*i2;

        float rad = 3.f * fmaxf(s0, fmaxf(s1, s2));

        float* g = G + k*GSTRIDE;
        g[0] = mu[k*3+0]; g[1] = mu[k*3+1]; g[2] = mu[k*3+2]; g[3] = rad*rad;
        g[4] = p00; g[5] = p01; g[6] = p02; g[7] = p11; g[8] = p12; g[9] = p22;
        g[10] = 0.f; g[11] = 0.f;
    }
    // B swizzle (4096 float2 entries)
    for (int i = threadIdx.x; i < 2*NSTEPS*32; i += 256) {
        int tile = i >> 11;           // / (64*32)
        int rem  = i & 2047;
        int step = rem >> 5;
        int lane = rem & 31;
        int kk   = step*4 + 2*(lane >> 4);
        int c    = tile*16 + (lane & 15);
        float f0 = (c < CDIM) ? F[kk*CDIM + c]     : 0.f;
        float f1 = (c < CDIM) ? F[(kk+1)*CDIM + c] : 0.f;
        Bsw[i*2+0] = f0;
        Bsw[i*2+1] = f1;
    }
}

// ---------------------------------------------------------------------------
// Gaussian evaluation from LDS-staged params
// ---------------------------------------------------------------------------
__device__ __forceinline__ float gauss_eval(const float* Gs, int k,
                                            float px, float py, float pz)
{
    const float4* g4 = reinterpret_cast<const float4*>(Gs + k*GSTRIDE);
    float4 a0 = g4[0];  // mux muy muz r2
    float4 a1 = g4[1];  // p00 p01 p02 p11
    float2 a2 = *reinterpret_cast<const float2*>(Gs + k*GSTRIDE + 8);  // p12 p22
    float dx = px - a0.x, dy = py - a0.y, dz = pz - a0.z;
    float d2 = dx*dx + dy*dy + dz*dz;
    float m  = a1.x*dx*dx + a1.w*dy*dy + a2.y*dz*dz
             + 2.f*(a1.y*dx*dy + a1.z*dx*dz + a2.x*dy*dz);
    float e = __expf(-0.5f*m);
    return (d2 < a0.w) ? e : 0.f;
}

// ---------------------------------------------------------------------------
// Kernel 2: 8 waves x 16 probes; gaussians in-register; WMMA f32 16x16x4
// ---------------------------------------------------------------------------
__global__ __launch_bounds__(256) void main_kernel(
    const float* __restrict__ probe_pos, const float* __restrict__ G,
    const float* __restrict__ Bsw, float* __restrict__ out_sh,
    float* __restrict__ out_w, int n)
{
    __shared__ __align__(16) float Gs[KG*GSTRIDE];       // 12 KB
    __shared__ __align__(16) float Bls[2*NSTEPS*32*2];   // 32 KB

    int t = threadIdx.x;
    for (int i = t; i < KG*GSTRIDE; i += 256)   Gs[i]  = G[i];
    for (int i = t; i < 2*NSTEPS*32*2; i += 256) Bls[i] = Bsw[i];
    __syncthreads();

    int wave = t >> 5, lane = t & 31;
    int p = lane & 15;        // probe within 16-row tile (== A-matrix M)
    int h = lane >> 4;        // k-parity half (A lanes 16-31 hold K=2,3)
    int n0 = blockIdx.x * TN;
    int probe = n0 + wave*16 + p;
    int pc = min(probe, n - 1);
    float px = probe_pos[pc*3+0];
    float py = probe_pos[pc*3+1];
    float pz = probe_pos[pc*3+2];

    // ---- pass 1: masked-gaussian sum over this lane's 128 k's ----
    float sum = 0.f;
    #pragma unroll 4
    for (int step = 0; step < NSTEPS; ++step) {
        int k = step*4 + 2*h;
        sum += gauss_eval(Gs, k,   px, py, pz);
        sum += gauss_eval(Gs, k+1, px, py, pz);
    }
    sum += __shfl_xor(sum, 16, 32);           // combine the two k-halves
    float invsum = 1.f / (sum + EPSF);

    // ---- pass 2: recompute g, write weights, accumulate WMMA ----
    v8f d0 = {}; v8f d1 = {};
    bool wvalid = probe < n;
    float* wrow = out_w + (size_t)probe * KG;

    #pragma unroll 2
    for (int step = 0; step < NSTEPS; ++step) {
        int k = step*4 + 2*h;
        float g0 = gauss_eval(Gs, k,   px, py, pz);
        float g1 = gauss_eval(Gs, k+1, px, py, pz);
        if (wvalid) {
            float2 wv = make_float2(g0*invsum, g1*invsum);
            *reinterpret_cast<float2*>(wrow + k) = wv;
        }
        v2f a; a.x = g0; a.y = g1;   // A fragment: this lane's (M=p, K=k..k+1)
        v2f b0 = *reinterpret_cast<const v2f*>(&Bls[(step*32 + lane)*2]);
        v2f b1 = *reinterpret_cast<const v2f*>(&Bls[2*NSTEPS*32 + (step*32 + lane)*2]);
        d0 = __builtin_amdgcn_wmma_f32_16x16x4_f32(false, a, false, b0,
                                                   (short)0, d0, false, false);
        d1 = __builtin_amdgcn_wmma_f32_16x16x4_f32(false, a, false, b1,
                                                   (short)0, d1, false, false);
    }

    // ---- scale rows by invsum (per probe) and store SH ----
    // D layout: VGPR j -> row M = j + 8*h, col = lane&15
    int rowbase = n0 + wave*16 + 8*h;
    #pragma unroll
    for (int j = 0; j < 8; ++j) {
        float sc = __shfl(invsum, j + 8*h, 32);  // lane (j+8h) owns probe (j+8h)
        int row = rowbase + j;
        if (row < n) {
            int c = lane & 15;
            out_sh[(size_t)row*CDIM + c] = d0[j] * sc;
            if (c < CDIM - 16)
                out_sh[(size_t)row*CDIM + 16 + c] = d1[j] * sc;
        }
    }
}

// ---------------------------------------------------------------------------
extern "C" void kernel_launch(void* const* d_in, const int* in_sizes, int n_in,
                              void* d_out, int out_size, void* d_ws, size_t ws_size,
                              hipStream_t stream) {
    const float* probe_pos = (const float*)d_in[0];  // [N,3]
    const float* mu        = (const float*)d_in[1];  // [K,3]
    const float* log_s     = (const float*)d_in[2];  // [K,3]
    const float* q         = (const float*)d_in[3];  // [K,4]
    const float* F         = (const float*)d_in[4];  // [K,27]
    int n = in_sizes[0] / 3;

    float* G   = (float*)d_ws;            // 256*12 floats = 12 KB
    float* Bsw = G + KG*GSTRIDE;          // 8192 floats   = 32 KB

    float* out_sh = (float*)d_out;                    // [N,27]
    float* out_w  = out_sh + (size_t)n * CDIM;        // [N,256]

    precompute_kernel<<<1, 256, 0, stream>>>(mu, log_s, q, F, G, Bsw);
    int blocks = (n + TN - 1) / TN;
    main_kernel<<<blocks, 256, 0, stream>>>(probe_pos, G, Bsw, out_sh, out_w, n);
}